// GatedChannelGroupAttention2d_77464030150917
// MI455X (gfx1250) — compile-verified
//
#include <hip/hip_runtime.h>
#include <hip/hip_bf16.h>

// ---------------------------------------------------------------------------
// GatedChannelGroupAttention2d, fused for MI455X (gfx1250).
//
// Algebraic collapse: every stage is a per-(b,channel) scalar multiply of an
// HxW plane plus fixed channel permutations; pooled means after gating are
// derivable from the ORIGINAL pooled means:
//   pooled2 = perm(gate1 * pooled1),  glob = mean(gate2 * pooled2)
// So:  out[b, cout(cin), :, :] = x[b, cin, :, :] * scale[b, cin]
// with scale = gate1[g,cpg] * gate2[g2,c2] * w[g2],
//      c' = cpg*16 + g, g2 = c'>>5, c2 = c'&31, cout = c2*16 + g2.
//
// Traffic: read x (128 MiB) in pool pass, re-read x from L2 (192 MB, primed
// by pool pass) + stream out (128 MiB NT stores)  =>  ~256 MiB HBM, ~11 us.
//
// Gate MLPs are real fp32 matmuls on V_WMMA_F32_16X16X4_F32: per batch,
// pooled[16x32] x gw1^T (8 K-steps), then h[16x8] x gw2^T (2 N-tiles x 2
// K-steps). One wave per batch; A and B fragments both fed from LDS
// (weights zero-padded in LDS so the WMMA loop has no predication).
// ---------------------------------------------------------------------------

typedef __attribute__((ext_vector_type(2))) float v2f;
typedef __attribute__((ext_vector_type(4))) float f4;
typedef __attribute__((ext_vector_type(8))) float v8f;

namespace {
constexpr int NB  = 16;    // batch
constexpr int NG  = 16;    // groups
constexpr int NC  = 32;    // channels per group
constexpr int NH  = 8;     // gate hidden
constexpr int NCH = 512;   // total channels
constexpr int NPX = 4096;  // H*W
}

// ---------------------------------------------------------------------------
// Kernel 1: pooled[b*512+c] = mean over HxW of x[b,c,:,:].
// One 256-thread block per plane; b128 loads, shuffle + LDS reduction.
// Default (RT) loads leave x resident in the 192MB L2 for the apply pass.
// ---------------------------------------------------------------------------
__global__ __launch_bounds__(256) void pool_kernel(const float* __restrict__ x,
                                                   float* __restrict__ pooled) {
  const int p = blockIdx.x;                       // plane index b*512 + c
  const f4* __restrict__ xin = (const f4*)(x + (size_t)p * NPX);
  const int t = threadIdx.x;

  float s = 0.0f;
#pragma unroll
  for (int i = 0; i < 4; ++i) {                   // 4 * 256 float4 = 4096 floats
    f4 v = xin[t + i * 256];
    s += (v.x + v.y) + (v.z + v.w);
  }
  // wave32 reduction
#pragma unroll
  for (int off = 16; off > 0; off >>= 1) s += __shfl_xor(s, off, 32);

  __shared__ float part[8];
  if ((t & 31) == 0) part[t >> 5] = s;
  __syncthreads();
  if (t == 0) {
    float tot = 0.0f;
#pragma unroll
    for (int i = 0; i < 8; ++i) tot += part[i];
    pooled[p] = tot * (1.0f / (float)NPX);
  }
}

// ---------------------------------------------------------------------------
// Kernel 2: one wave (32 lanes) per batch computes scale[b, cin].
// WMMA f32 16x16x4 fragment layouts (ISA 7.12.2):
//   A 16x4 : lanes 0-15 rows M, VGPR{0,1} = K{k0,k0+1}; lanes 16-31 = K{k0+2,k0+3}
//   B 4x16 : lanes 0-15 cols N, VGPR{0,1} = K{k0,k0+1}; lanes 16-31 = K{k0+2,k0+3}
//   C 16x16: VGPR r -> M = r (lanes 0-15) / r+8 (lanes 16-31), N = lane%16
// ---------------------------------------------------------------------------
__global__ __launch_bounds__(32) void gate_scale_kernel(
    const float* __restrict__ pooled,
    const float* __restrict__ gw1, const float* __restrict__ gb1,
    const float* __restrict__ gw2, const float* __restrict__ gb2,
    const float* __restrict__ cw1, const float* __restrict__ cb1,
    const float* __restrict__ cw2, const float* __restrict__ cb2,
    float* __restrict__ scale) {
  __shared__ float P [NG][NC];   // pooled1
  __shared__ float Hh[NG][NH];   // hidden (relu)
  __shared__ float G1[NG][NC];   // gate of pass 1
  __shared__ float P2[NG][NC];   // shuffled gated pooled
  __shared__ float G2[NG][NC];   // gate of pass 2
  __shared__ float W1[NG][NC];   // gw1 zero-padded to 16 output rows
  __shared__ float W2[NC][NH];   // gw2
  __shared__ float GLB[NG];
  __shared__ float WV[NG];

  const int b    = blockIdx.x;
  const int lane = threadIdx.x;     // 0..31
  const int half = lane >> 4;       // which K-pair of the fragment
  const int l16  = lane & 15;       // row (A/C) or col (B/C) within tile

  // one-time staging: pooled1[b] and zero-padded weights -> LDS
  for (int i = 0; i < 16; ++i) {
    int idx = i * 32 + lane;                       // 0..511
    P[idx >> 5][idx & 31] = pooled[b * NCH + idx];
    int o = idx >> 5, c = idx & 31;
    W1[o][c] = (o < NH) ? gw1[o * NC + c] : 0.0f;  // pad rows 8..15 with 0
  }
  for (int i = 0; i < 8; ++i) {
    int idx = i * 32 + lane;                       // 0..255
    W2[idx >> 3][idx & 7] = gw2[idx];
  }
  __syncthreads();

  // One full gate pass:  Gout = sigmoid(gw2 @ relu(gw1 @ Pin + gb1) + gb2)
  auto gate_pass = [&](float (*Pin)[NC], float (*Gout)[NC]) {
    // ---- layer 1: [16x32] x [32x16(padded)] via 8 chained WMMAs ----
    v8f acc = {0, 0, 0, 0, 0, 0, 0, 0};
#pragma unroll
    for (int k0 = 0; k0 < NC; k0 += 4) {
      const int kk = k0 + 2 * half;
      v2f a, bw;
      a.x  = Pin[l16][kk + 0];
      a.y  = Pin[l16][kk + 1];
      bw.x = W1[l16][kk + 0];                      // B[k][n] = gw1[n][k], padded
      bw.y = W1[l16][kk + 1];
      acc = __builtin_amdgcn_wmma_f32_16x16x4_f32(false, a, false, bw,
                                                  (short)0, acc, false, false);
    }
    if (l16 < NH) {
      const float bias = gb1[l16];
#pragma unroll
      for (int r = 0; r < 8; ++r) {
        const int g = r + half * 8;
        const float v = acc[r] + bias;
        Hh[g][l16] = v > 0.0f ? v : 0.0f;
      }
    }
    __syncthreads();

    // ---- layer 2: [16x8] x [8x32] -> two N-tiles, K=8 (2 WMMAs each) ----
#pragma unroll
    for (int tile = 0; tile < 2; ++tile) {
      const int cb = tile * 16;
      v8f acc2 = {0, 0, 0, 0, 0, 0, 0, 0};
#pragma unroll
      for (int k0 = 0; k0 < NH; k0 += 4) {
        const int kk = k0 + 2 * half;
        v2f a, bw;
        a.x  = Hh[l16][kk + 0];
        a.y  = Hh[l16][kk + 1];
        bw.x = W2[cb + l16][kk + 0];               // B[k][n] = gw2[n][k]
        bw.y = W2[cb + l16][kk + 1];
        acc2 = __builtin_amdgcn_wmma_f32_16x16x4_f32(false, a, false, bw,
                                                     (short)0, acc2, false, false);
      }
      const int c = cb + l16;
      const float bias = gb2[c];
#pragma unroll
      for (int r = 0; r < 8; ++r) {
        const int g = r + half * 8;
        const float v = acc2[r] + bias;
        Gout[g][c] = 1.0f / (1.0f + __expf(-v));
      }
    }
    __syncthreads();
  };

  // pass 1
  gate_pass(P, G1);

  // analytic shuffle: P2[g2][c2] = G1[g][cpg] * P[g][cpg], c' = cpg*16+g
  {
    const int cpg = lane;
    for (int g = 0; g < NG; ++g) {
      const int cp = cpg * NG + g;
      P2[cp >> 5][cp & 31] = G1[g][cpg] * P[g][cpg];
    }
  }
  __syncthreads();

  // pass 2
  gate_pass(P2, G2);

  // glob[g2] = mean_c2 (G2 * P2); tiny cross-gate MLP 16->8->16 in VALU
  if (lane < NG) {
    float s = 0.0f;
    for (int c = 0; c < NC; ++c) s += G2[lane][c] * P2[lane][c];
    GLB[lane] = s * (1.0f / (float)NC);
  }
  __syncthreads();
  if (lane < NG) {
    float h2[NH];
#pragma unroll
    for (int j = 0; j < NH; ++j) {
      float s = cb1[j];
      for (int g = 0; g < NG; ++g) s += cw1[j * NG + g] * GLB[g];
      h2[j] = s > 0.0f ? s : 0.0f;
    }
    float s = cb2[lane];
#pragma unroll
    for (int j = 0; j < NH; ++j) s += cw2[lane * NH + j] * h2[j];
    WV[lane] = 1.0f / (1.0f + __expf(-s));
  }
  __syncthreads();

  // scale per INPUT channel cin = g*32 + cpg
  {
    const int cpg = lane;
    for (int g = 0; g < NG; ++g) {
      const int cp = cpg * NG + g;
      const int g2 = cp >> 5, c2 = cp & 31;
      scale[b * NCH + g * NC + cpg] = G1[g][cpg] * G2[g2][c2] * WV[g2];
    }
  }
}

// ---------------------------------------------------------------------------
// Kernel 3: out[b, cout(cin), :, :] = x[b, cin, :, :] * scale[b, cin].
// One block per input plane. x loads hit L2 (primed by pool_kernel);
// output streamed with non-temporal b128 stores so it doesn't evict x.
// ---------------------------------------------------------------------------
__global__ __launch_bounds__(256) void apply_kernel(const float* __restrict__ x,
                                                    const float* __restrict__ scale,
                                                    float* __restrict__ out) {
  const int p   = blockIdx.x;          // b*512 + cin
  const int b   = p >> 9;
  const int cin = p & 511;
  const int g   = cin >> 5, cpg = cin & 31;
  const int cp  = cpg * NG + g;
  const int g2  = cp >> 5, c2 = cp & 31;
  const int cout = c2 * NG + g2;

  const float s = scale[p];            // block-uniform -> scalar load
  const f4* __restrict__ xin = (const f4*)(x + (size_t)p * NPX);
  f4* __restrict__ op = (f4*)(out + ((size_t)b * NCH + cout) * NPX);

  const int t = threadIdx.x;
#pragma unroll
  for (int i = 0; i < 4; ++i) {
    f4 v = xin[t + i * 256];
    v *= s;
    __builtin_nontemporal_store(v, &op[t + i * 256]);
  }
}

extern "C" void kernel_launch(void* const* d_in, const int* in_sizes, int n_in,
                              void* d_out, int out_size, void* d_ws, size_t ws_size,
                              hipStream_t stream) {
  const float* x   = (const float*)d_in[0];
  const float* gw1 = (const float*)d_in[1];
  const float* gb1 = (const float*)d_in[2];
  const float* gw2 = (const float*)d_in[3];
  const float* gb2 = (const float*)d_in[4];
  const float* cw1 = (const float*)d_in[5];
  const float* cb1 = (const float*)d_in[6];
  const float* cw2 = (const float*)d_in[7];
  const float* cb2 = (const float*)d_in[8];

  // workspace: pooled[16*512] then scale[16*512]  (64 KB total)
  float* pooled = (float*)d_ws;
  float* scale  = pooled + NB * NCH;

  pool_kernel<<<NB * NCH, 256, 0, stream>>>(x, pooled);
  gate_scale_kernel<<<NB, 32, 0, stream>>>(pooled, gw1, gb1, gw2, gb2,
                                           cw1, cb1, cw2, cb2, scale);
  apply_kernel<<<NB * NCH, 256, 0, stream>>>(x, scale, (float*)d_out);
}